// PLRNNet_8924942041578
// MI455X (gfx1250) — compile-verified
//
#include <hip/hip_runtime.h>

typedef __attribute__((ext_vector_type(2))) float v2f;
typedef __attribute__((ext_vector_type(4))) float v4f;
typedef __attribute__((ext_vector_type(8))) float v8f;

#define DS 16    // subsets
#define DC 32    // contexts
#define DL 1024  // length
#define DF 64    // latent features
#define DN 128   // readout neurons

// -------------------------------------------------------------------------
// Kernel 1: PLRNN scan.  One wave (32 lanes) per 16 sequences (S*C/16 = 32
// waves total).  State Z (16 seq x 64 feat) kept in WMMA C/D layout:
//   acc[nt][r] = Z[m = r + 8*hi][f = 16*nt + lo],  lo = lane&15, hi = lane>>4
// Step:  Znew = A (.) Z + relu(Z) @ W^T + h  via 4 x (16-deep K chain) of
// v_wmma_f32_16x16x4_f32, with C pre-seeded to A(.)Z + h.
// W^T B-operand tiles are register-resident (loop-invariant over 1023 steps).
// -------------------------------------------------------------------------
__global__ __launch_bounds__(32) void plrnn_scan(
    const float* __restrict__ z0, const float* __restrict__ Adiag,
    const float* __restrict__ W, const float* __restrict__ h,
    float* __restrict__ zout) {
  __shared__ float zs[16 * 66];  // relu(Z) staging, padded stride
  const int l = threadIdx.x;
  const int lo = l & 15;
  const int hi = l >> 4;
  const int g0 = blockIdx.x * 16;  // flattened (s*C + c) base

  // B-operand tiles of W^T: B[kt][nt] element (kr, nc) = W[16*nt+nc][4*kt+kr]
  // lane holds (kr = 2*hi, 2*hi+1) for nc = lo  -> consecutive floats in W row.
  v2f bw[16][4];
#pragma unroll
  for (int kt = 0; kt < 16; ++kt)
#pragma unroll
    for (int nt = 0; nt < 4; ++nt)
      bw[kt][nt] =
          *reinterpret_cast<const v2f*>(W + (nt * 16 + lo) * DF + kt * 4 + hi * 2);

  float ad[4], hv[4];
#pragma unroll
  for (int nt = 0; nt < 4; ++nt) {
    ad[nt] = Adiag[nt * 16 + lo];
    hv[nt] = h[nt * 16 + lo];
  }

  // load z0 into C/D-layout accumulators
  v8f acc[4];
#pragma unroll
  for (int nt = 0; nt < 4; ++nt)
#pragma unroll
    for (int r = 0; r < 8; ++r)
      acc[nt][r] = z0[(size_t)(g0 + r + 8 * hi) * DF + nt * 16 + lo];

  for (int t = 0; t < DL; ++t) {
    // emit z at time t  (z layout: ((s*C+c)*L + t)*F + f)
#pragma unroll
    for (int nt = 0; nt < 4; ++nt)
#pragma unroll
      for (int r = 0; r < 8; ++r)
        zout[((size_t)(g0 + r + 8 * hi) * DL + t) * DF + nt * 16 + lo] =
            acc[nt][r];
    if (t == DL - 1) break;

    // stage relu(Z) row-major into LDS
#pragma unroll
    for (int nt = 0; nt < 4; ++nt)
#pragma unroll
      for (int r = 0; r < 8; ++r)
        zs[(r + 8 * hi) * 66 + nt * 16 + lo] = fmaxf(acc[nt][r], 0.0f);
    __syncthreads();

    // seed C with A (.) Z + h
    v8f c[4];
#pragma unroll
    for (int nt = 0; nt < 4; ++nt)
#pragma unroll
      for (int r = 0; r < 8; ++r) c[nt][r] = ad[nt] * acc[nt][r] + hv[nt];

    // K chain: A-operand (16x4 f32 layout): row = lo, K = 4*kt + 2*hi + {0,1}
#pragma unroll
    for (int kt = 0; kt < 16; ++kt) {
      v2f a = *reinterpret_cast<const v2f*>(&zs[lo * 66 + kt * 4 + hi * 2]);
#pragma unroll
      for (int nt = 0; nt < 4; ++nt)
        c[nt] = __builtin_amdgcn_wmma_f32_16x16x4_f32(
            false, a, false, bw[kt][nt], (short)0, c[nt], false, false);
    }
    __syncthreads();  // WAR protection on zs before next step's stores
#pragma unroll
    for (int nt = 0; nt < 4; ++nt) acc[nt] = c[nt];
  }
}

// -------------------------------------------------------------------------
// Kernel 2: readout  x[s, m, n] = sum_f z[s, m, f] * obs_W[s, f, n]
// (m = c*L + t flattened, 32768 rows per subset).
// 128-thread blocks (4 waves).  obs_W[s] staged transposed in LDS so the
// B operand is a b64 LDS load of consecutive K; each wave stages its 16x64
// z tile via coalesced b128 loads, then 16(K) x 8(N) WMMAs (8 interleaved
// accumulation chains).
// -------------------------------------------------------------------------
__global__ __launch_bounds__(128) void plrnn_readout(
    const float* __restrict__ zin, const float* __restrict__ obsW,
    float* __restrict__ x) {
  __shared__ float wt[128 * 66];     // wt[n][k] = obs_W[s][k][n]
  __shared__ float zb[4 * 16 * 68];  // per-wave 16x64 z tile, 16B-aligned rows
  const int tid = threadIdx.x;
  const int w = tid >> 5;
  const int l = tid & 31;
  const int lo = l & 15;
  const int hi = l >> 4;
  const int s = blockIdx.x >> 9;        // 512 blocks per subset
  const int mchunk = blockIdx.x & 511;  // 4 m-tiles per block (1 per wave)
  const size_t rows = (size_t)DC * DL;  // 32768
  const size_t R0 = ((size_t)mchunk * 4 + w) * 16;

  // cooperative transpose-stage of obs_W[s] (64x128 -> wt[n][k])
  const float* Ws = obsW + (size_t)s * DF * DN;
  for (int i = 0; i < 64; ++i) {
    int idx = tid + 128 * i;  // coalesced read: consecutive n within a k row
    int k = idx >> 7, n = idx & 127;
    wt[n * 66 + k] = Ws[idx];
  }

  // per-wave z tile stage: 16 rows x 64 floats, b128 coalesced
  const float* zsrc = zin + ((size_t)s * rows + R0) * DF;
  float* zbw = zb + w * (16 * 68);
#pragma unroll
  for (int i = 0; i < 8; ++i) {
    int off = l * 4 + i * 128;
    int row = off >> 6, col = off & 63;
    *reinterpret_cast<v4f*>(&zbw[row * 68 + col]) =
        *reinterpret_cast<const v4f*>(zsrc + off);
  }
  __syncthreads();

  // A operands: row = lo, K = 4*kt + 2*hi + {0,1}
  v2f a[16];
#pragma unroll
  for (int kt = 0; kt < 16; ++kt)
    a[kt] = *reinterpret_cast<const v2f*>(&zbw[lo * 68 + kt * 4 + hi * 2]);

  v8f acc[8];
#pragma unroll
  for (int nt = 0; nt < 8; ++nt) acc[nt] = (v8f){};

#pragma unroll
  for (int kt = 0; kt < 16; ++kt) {
#pragma unroll
    for (int nt = 0; nt < 8; ++nt) {
      v2f b = *reinterpret_cast<const v2f*>(
          &wt[(nt * 16 + lo) * 66 + kt * 4 + hi * 2]);
      acc[nt] = __builtin_amdgcn_wmma_f32_16x16x4_f32(
          false, a[kt], false, b, (short)0, acc[nt], false, false);
    }
  }

  // store 16x128 x strip: x layout (s*rows + m)*N + n
  float* xo = x + ((size_t)s * rows + R0) * DN;
#pragma unroll
  for (int nt = 0; nt < 8; ++nt)
#pragma unroll
    for (int r = 0; r < 8; ++r)
      xo[(size_t)(r + 8 * hi) * DN + nt * 16 + lo] = acc[nt][r];
}

extern "C" void kernel_launch(void* const* d_in, const int* in_sizes, int n_in,
                              void* d_out, int out_size, void* d_ws,
                              size_t ws_size, hipStream_t stream) {
  const float* z0 = (const float*)d_in[0];    // (S, C, F)
  const float* A = (const float*)d_in[1];     // (F,)
  const float* W = (const float*)d_in[2];     // (F, F)
  const float* h = (const float*)d_in[3];     // (F,)
  const float* obsW = (const float*)d_in[4];  // (S, F, N)

  float* x = (float*)d_out;                        // (S, C, L, N)
  float* zout = x + (size_t)DS * DC * DL * DN;     // (S, C, L, F)

  plrnn_scan<<<(DS * DC) / 16, 32, 0, stream>>>(z0, A, W, h, zout);
  plrnn_readout<<<DS * 512, 128, 0, stream>>>(zout, obsW, x);
}